// ProteinFeatures_3607772528734
// MI455X (gfx1250) — compile-verified
//
#include <hip/hip_runtime.h>
#include <hip/hip_bf16.h>
#include <math.h>

typedef __attribute__((ext_vector_type(2))) float v2f;
typedef __attribute__((ext_vector_type(8))) float v8f;

#define Bn 8
#define Ln 2048
#define Kn 30
#define NRBF 16
#define EFEAT 128
#define FDIM 416          // 16 positional + 25*16 RBF
#define MAXREL 32
#define FP 420            // feat LDS row pitch (floats), 16B-multiple rows
#define OPITCH 132        // out LDS row pitch

// atom order: N=0, Ca=1, C=2, O=3, Cb=4; pair 0 = (Ca,Ca) from D_neighbors
__constant__ int PA_tab[25] = {1, 0,2,3,4, 1,1,1,1, 0,0,0, 4,4, 3, 0,2,3,4, 2,3,4, 2,3, 2};
__constant__ int PB_tab[25] = {1, 0,2,3,4, 0,2,3,4, 2,3,4, 2,3, 2, 1,1,1,1, 0,0,0, 4,4, 3};

// ---------------- kernel 0: pack atoms + virtual Cb ----------------
__global__ void prep_atoms(const float* __restrict__ X, float* __restrict__ atoms) {
    int idx = blockIdx.x * blockDim.x + threadIdx.x;
    if (idx >= Bn * Ln) return;
    const float* x = X + (size_t)idx * 12;
    float Nx=x[0],Ny=x[1],Nz=x[2];
    float Ax=x[3],Ay=x[4],Az=x[5];
    float Cx=x[6],Cy=x[7],Cz=x[8];
    float Ox=x[9],Oy=x[10],Oz=x[11];
    float bx=Ax-Nx, by=Ay-Ny, bz=Az-Nz;        // b = Ca - N
    float cx=Cx-Ax, cy=Cy-Ay, cz=Cz-Az;        // c = C - Ca
    float ax = by*cz - bz*cy;                   // a = cross(b,c)
    float ay = bz*cx - bx*cz;
    float az = bx*cy - by*cx;
    float cbx = -0.58273431f*ax + 0.56802827f*bx - 0.54067466f*cx + Ax;
    float cby = -0.58273431f*ay + 0.56802827f*by - 0.54067466f*cy + Ay;
    float cbz = -0.58273431f*az + 0.56802827f*bz - 0.54067466f*cz + Az;
    float* o = atoms + (size_t)idx * 15;
    o[0]=Nx; o[1]=Ny; o[2]=Nz;
    o[3]=Ax; o[4]=Ay; o[5]=Az;
    o[6]=Cx; o[7]=Cy; o[8]=Cz;
    o[9]=Ox; o[10]=Oy; o[11]=Oz;
    o[12]=cbx; o[13]=cby; o[14]=cbz;
}

// ---------------- kernel 1: masked distances + top-K (one wave32 per row) ----
__global__ __launch_bounds__(128) void topk_kernel(
    const float* __restrict__ atoms, const float* __restrict__ mask,
    float* __restrict__ Dnb, int* __restrict__ Eidx_ws, int* __restrict__ Eidx_out)
{
    __shared__ float drow[4][Ln];                 // 32 KB
    int wave = threadIdx.x >> 5;
    int lane = threadIdx.x & 31;
    int row  = blockIdx.x * 4 + wave;             // (b,i) flat
    int b = row / Ln;
    float* dr = drow[wave];

    const float* cai = atoms + ((size_t)row * 5 + 1) * 3;
    float cax = cai[0], cay = cai[1], caz = cai[2];
    float mi = mask[row];
    const float* abase = atoms + (size_t)b * Ln * 15;
    const float* mbase = mask + (size_t)b * Ln;

    // pass A: D = m_i*m_j*sqrt(|dCa|^2 + 1e-6); track masked row max
    float dmax = 0.0f;
    for (int t = 0; t < Ln / 32; ++t) {
        int j = lane + t * 32;
        const float* cj = abase + (size_t)j * 15 + 3;
        float dx = cax - cj[0], dy = cay - cj[1], dz = caz - cj[2];
        float d  = sqrtf(dx*dx + dy*dy + dz*dz + 1e-6f);
        float md = mi * mbase[j] * d;
        dr[j] = md;
        dmax = fmaxf(dmax, md);
    }
    for (int off = 16; off; off >>= 1)
        dmax = fmaxf(dmax, __shfl_xor(dmax, off, 32));

    // pass B: D_adjust = D + (1 - m2d) * D_max
    for (int t = 0; t < Ln / 32; ++t) {
        int j = lane + t * 32;
        float mm = mi * mbase[j];
        dr[j] += (1.0f - mm) * dmax;
    }

    // K rounds of argmin (value then lowest index, matching top_k tie-break)
    for (int it = 0; it < Kn; ++it) {
        float bv = 3.4e38f; int bi = Ln;
        for (int t = 0; t < Ln / 32; ++t) {
            int j = lane + t * 32;
            float v = dr[j];
            if (v < bv) { bv = v; bi = j; }
        }
        for (int off = 16; off; off >>= 1) {
            float ov = __shfl_xor(bv, off, 32);
            int   oi = __shfl_xor(bi, off, 32);
            if (ov < bv || (ov == bv && oi < bi)) { bv = ov; bi = oi; }
        }
        if (lane == 0) {
            Dnb[(size_t)row * Kn + it]      = bv;
            Eidx_ws[(size_t)row * Kn + it]  = bi;
            Eidx_out[(size_t)row * Kn + it] = bi;
            dr[bi] = 3.4e38f;
        }
    }
}

// ---------------- kernel 2: fused features + WMMA GEMM + LayerNorm ----------
__global__ __launch_bounds__(256) void edge_kernel(
    const float* __restrict__ atoms, const int* __restrict__ Eidx,
    const float* __restrict__ Dnb,
    const int* __restrict__ residx, const int* __restrict__ chains,
    const float* __restrict__ pos_w, const float* __restrict__ pos_b,
    const float* __restrict__ edge_w,
    const float* __restrict__ gamma, const float* __restrict__ beta,
    float* __restrict__ Eout)
{
    __shared__ __align__(16) float feat[16 * FP];      // 16 edges x 416 features
    __shared__ __align__(16) float outb[16 * OPITCH];  // 16 edges x 128 outputs
    __shared__ int   jmeta[16], dmeta[16], bmeta[16], imeta[16];
    __shared__ float dnb0[16];
    __shared__ float mu_s[16], rstd_s[16];

    const int tile = blockIdx.x;
    const int tid  = threadIdx.x;

    // --- per-edge metadata ---
    if (tid < 16) {
        int e  = tile * 16 + tid;
        int b  = e / (Ln * Kn);
        int r  = e - b * (Ln * Kn);
        int i  = r / Kn;
        int kk = r - i * Kn;
        int ri_flat = b * Ln + i;
        int j  = Eidx[(size_t)ri_flat * Kn + kk];
        bmeta[tid] = b; imeta[tid] = i; jmeta[tid] = j;
        int off = residx[ri_flat] - residx[b * Ln + j] + MAXREL;
        off = off < 0 ? 0 : (off > 2 * MAXREL ? 2 * MAXREL : off);
        int eq = (chains[ri_flat] == chains[b * Ln + j]);
        dmeta[tid] = eq ? off : (2 * MAXREL + 1);
        dnb0[tid]  = Dnb[(size_t)ri_flat * Kn + kk];
    }
    __syncthreads();

    // --- positional features (one-hot @ pos_w.T + pos_b == row lookup) ---
    {
        int s = tid >> 4, f = tid & 15;
        feat[s * FP + f] = pos_w[f * (2 * MAXREL + 2) + dmeta[s]] + pos_b[f];
    }
    // --- 25 RBF blocks per edge: 400 (edge,pair) combos over 256 threads ---
    for (int idx = tid; idx < 16 * 25; idx += 256) {
        int s = idx / 25;
        int p = idx - s * 25;
        float dist;
        if (p == 0) {
            dist = dnb0[s];                        // masked D_neighbors
        } else {
            int b = bmeta[s], i = imeta[s], j = jmeta[s];
            const float* pa = atoms + ((size_t)(b * Ln + i) * 5 + PA_tab[p]) * 3;
            const float* pb = atoms + ((size_t)(b * Ln + j) * 5 + PB_tab[p]) * 3;
            float dx = pa[0] - pb[0], dy = pa[1] - pb[1], dz = pa[2] - pb[2];
            dist = sqrtf(dx*dx + dy*dy + dz*dz + 1e-6f);
        }
        float* fr = &feat[s * FP + 16 + p * 16];
        #pragma unroll
        for (int r = 0; r < NRBF; ++r) {
            float mu = 2.0f + (20.0f / 15.0f) * (float)r;   // linspace(2,22,16)
            float t  = (dist - mu) * 0.8f;                   // / sigma(=1.25)
            fr[r] = __expf(-t * t);
        }
    }
    __syncthreads();

    // --- fp32 WMMA GEMM: (16 x 416) @ (416 x 128), wave w owns cols [16w,16w+16) ---
    const int wave = tid >> 5;
    const int lane = tid & 31;
    const int n0   = wave * 16;
    const int ln15 = lane & 15;
    const int kp   = (lane >> 4) * 2;   // lanes 0-15 -> K {0,1}; lanes 16-31 -> K {2,3}
    const float* wrow = edge_w + (size_t)(n0 + ln15) * FDIM + kp;  // B[k][n] frag base
    const float* arow = feat + ln15 * FP + kp;                     // A[m][k] frag base

    v8f acc = {0.f,0.f,0.f,0.f,0.f,0.f,0.f,0.f};
    #pragma unroll 4
    for (int k0 = 0; k0 < FDIM; k0 += 4) {
        v2f a = *(const v2f*)(arow + k0);   // LDS, 8B aligned
        v2f w = *(const v2f*)(wrow + k0);   // global (L2-resident weights)
        acc = __builtin_amdgcn_wmma_f32_16x16x4_f32(
                  false, a, false, w, (short)0, acc, false, false);
    }

    // C/D layout: VGPR r, lane l -> M = r + 8*(l>>4), N = l&15
    #pragma unroll
    for (int r = 0; r < 8; ++r) {
        int row = r + (lane >> 4) * 8;
        outb[row * OPITCH + n0 + ln15] = acc[r];
    }
    __syncthreads();

    // --- LayerNorm over 128 outputs per edge ---
    if (tid < 16) {
        float s1 = 0.f, s2 = 0.f;
        const float* rw = &outb[tid * OPITCH];
        for (int c = 0; c < EFEAT; ++c) { float v = rw[c]; s1 += v; s2 += v * v; }
        float m   = s1 * (1.0f / EFEAT);
        float var = s2 * (1.0f / EFEAT) - m * m;
        mu_s[tid]   = m;
        rstd_s[tid] = rsqrtf(var + 1e-5f);
    }
    __syncthreads();
    {
        int s = tid >> 4, chunk = tid & 15;
        float m = mu_s[s], rs = rstd_s[s];
        size_t e = (size_t)tile * 16 + s;
        float*       dst = Eout + e * EFEAT + chunk * 8;
        const float* src = &outb[s * OPITCH + chunk * 8];
        #pragma unroll
        for (int c = 0; c < 8; ++c) {
            int col = chunk * 8 + c;
            dst[c] = (src[c] - m) * rs * gamma[col] + beta[col];
        }
    }
}

// ---------------- host launcher ----------------
extern "C" void kernel_launch(void* const* d_in, const int* in_sizes, int n_in,
                              void* d_out, int out_size, void* d_ws, size_t ws_size,
                              hipStream_t stream) {
    const float* X      = (const float*)d_in[0];
    const float* mask   = (const float*)d_in[1];
    const int*   residx = (const int*)d_in[2];
    const int*   chains = (const int*)d_in[3];
    const float* pos_w  = (const float*)d_in[4];
    const float* pos_b  = (const float*)d_in[5];
    const float* edge_w = (const float*)d_in[6];
    const float* gamma  = (const float*)d_in[7];
    const float* beta   = (const float*)d_in[8];

    float* Eout     = (float*)d_out;
    int*   Eidx_out = (int*)d_out + (size_t)Bn * Ln * Kn * EFEAT;

    // workspace: atoms (B*L*15 f32) | Dnb (B*L*K f32) | Eidx (B*L*K i32)
    float* atoms   = (float*)d_ws;
    float* Dnb     = atoms + (size_t)Bn * Ln * 15;
    int*   Eidx_ws = (int*)(Dnb + (size_t)Bn * Ln * Kn);

    prep_atoms<<<(Bn * Ln + 255) / 256, 256, 0, stream>>>(X, atoms);
    topk_kernel<<<(Bn * Ln) / 4, 128, 0, stream>>>(atoms, mask, Dnb, Eidx_ws, Eidx_out);
    edge_kernel<<<(Bn * Ln * Kn) / 16, 256, 0, stream>>>(
        atoms, Eidx_ws, Dnb, residx, chains, pos_w, pos_b, edge_w, gamma, beta, Eout);
}